// attention_learn_85882166050971
// MI455X (gfx1250) — compile-verified
//
#include <hip/hip_runtime.h>
#include <hip/hip_bf16.h>
#include <cstddef>

// Problem constants (B=1, C=64, N=M=768)
#define WG 256
constexpr int C64  = 64;
constexpr int NPTS = 768;
constexpr int NT16 = NPTS / 16;   // 48 n-tiles

typedef __attribute__((ext_vector_type(2))) float v2f;
typedef __attribute__((ext_vector_type(8))) float v8f;

#if defined(__gfx1250__)
#  if __has_builtin(__builtin_amdgcn_wmma_f32_16x16x4_f32)
#    define USE_WMMA_F32 1
#  endif
#endif

// ---------------------------------------------------------------------------
// Generic fp32 WMMA GEMM: C[M,N] = A[M,K] * B[K,N] (+bias[row]) (+resid)
// One wave (32 threads) computes a 64-row x 16-col strip of C.
//   grid.x = N/16, grid.y = M/64.
// Compile-time specialized on BT / HAS_BIAS / HAS_RES so the K-loop is
// branch-free (loads + v_wmma_f32_16x16x4_f32 only).
// BT=1 means B is accessed transposed: B[k][col] = Bsrc[col*ldb + k].
// K must be a multiple of 4.
// ---------------------------------------------------------------------------
template <int BT, bool HAS_BIAS, bool HAS_RES>
__global__ void __launch_bounds__(32) wmma_gemm_kernel(
    const float* __restrict__ A, int lda,
    const float* __restrict__ B, int ldb,
    const float* __restrict__ bias,
    const float* __restrict__ resid, int ldr,
    float* __restrict__ Cout, int ldc, int K)
{
    const int lane  = threadIdx.x & 31;
    const int l16   = lane & 15;
    const int lhi   = (lane >> 4) & 1;
    const int ntile = blockIdx.x << 4;
    const int mbase = blockIdx.y << 6;

    // Per-lane base pointers (hoisted out of the K-loop).
    const float* a0 = A + (size_t)(mbase +  0 + l16) * lda + 2 * lhi;
    const float* a1 = A + (size_t)(mbase + 16 + l16) * lda + 2 * lhi;
    const float* a2 = A + (size_t)(mbase + 32 + l16) * lda + 2 * lhi;
    const float* a3 = A + (size_t)(mbase + 48 + l16) * lda + 2 * lhi;
    const float* bp = BT ? (B + (size_t)(ntile + l16) * ldb + 2 * lhi)
                         : (B + (size_t)(2 * lhi) * ldb + ntile + l16);

    float accf[4][8];
#ifdef USE_WMMA_F32
    v8f acc[4];
    for (int t = 0; t < 4; ++t)
        for (int r = 0; r < 8; ++r) acc[t][r] = 0.0f;

    for (int kk = 0; kk < K; kk += 4) {
        v2f bfrag;
        if (BT) { bfrag.x = bp[kk]; bfrag.y = bp[kk + 1]; }
        else    { bfrag.x = bp[(size_t)kk * ldb]; bfrag.y = bp[(size_t)(kk + 1) * ldb]; }
        v2f af;
        af.x = a0[kk]; af.y = a0[kk + 1];
        acc[0] = __builtin_amdgcn_wmma_f32_16x16x4_f32(false, af, false, bfrag, (short)0, acc[0], false, false);
        af.x = a1[kk]; af.y = a1[kk + 1];
        acc[1] = __builtin_amdgcn_wmma_f32_16x16x4_f32(false, af, false, bfrag, (short)0, acc[1], false, false);
        af.x = a2[kk]; af.y = a2[kk + 1];
        acc[2] = __builtin_amdgcn_wmma_f32_16x16x4_f32(false, af, false, bfrag, (short)0, acc[2], false, false);
        af.x = a3[kk]; af.y = a3[kk + 1];
        acc[3] = __builtin_amdgcn_wmma_f32_16x16x4_f32(false, af, false, bfrag, (short)0, acc[3], false, false);
    }
    for (int t = 0; t < 4; ++t)
        for (int r = 0; r < 8; ++r) accf[t][r] = acc[t][r];
#else
    // Scalar fallback with the identical output mapping (also parsed on host).
    for (int t = 0; t < 4; ++t) {
        for (int r = 0; r < 8; ++r) {
            const int row = mbase + t * 16 + r + 8 * lhi;
            const int col = ntile + l16;
            float s = 0.0f;
            for (int k2 = 0; k2 < K; ++k2)
                s += A[(size_t)row * lda + k2] *
                     (BT ? B[(size_t)col * ldb + k2] : B[(size_t)k2 * ldb + col]);
            accf[t][r] = s;
        }
    }
#endif
    // Epilogue per ISA C/D layout: VGPR r -> M = r (lanes 0-15) / r+8 (lanes 16-31)
    for (int t = 0; t < 4; ++t) {
        for (int r = 0; r < 8; ++r) {
            const int row = mbase + t * 16 + r + 8 * lhi;
            const int col = ntile + l16;
            float v = accf[t][r];
            if (HAS_BIAS) v += bias[row];
            if (HAS_RES)  v += resid[(size_t)row * ldr + col];
            Cout[(size_t)row * ldc + col] = v;
        }
    }
}

// ---------------------------------------------------------------------------
// Stage-1 norms: inorm2d + bnorm2d(sep_bn1) + relu collapse to per-channel
// 1-D affine of q / k (broadcast channels keep their 1-D stats exactly).
// blockIdx.x = channel c in [0,128): c<64 -> q-row, else k-row.
// ---------------------------------------------------------------------------
__global__ void __launch_bounds__(WG) stage1_norm_kernel(
    const float* __restrict__ q, const float* __restrict__ kx,
    const float* __restrict__ g, const float* __restrict__ b,
    float* __restrict__ Aq, float* __restrict__ Bk)
{
    const int c   = blockIdx.x;
    const int tid = threadIdx.x;
    const float* src = (c < C64) ? (q  + (size_t)c        * NPTS)
                                 : (kx + (size_t)(c - C64) * NPTS);
    float*       dst = (c < C64) ? (Aq + (size_t)c        * NPTS)
                                 : (Bk + (size_t)(c - C64) * NPTS);
    __shared__ float rs[WG], rq[WG];
    float s = 0.f, sq = 0.f;
    for (int i = tid; i < NPTS; i += WG) { float x = src[i]; s += x; sq += x * x; }
    rs[tid] = s; rq[tid] = sq; __syncthreads();
    for (int off = WG / 2; off > 0; off >>= 1) {
        if (tid < off) { rs[tid] += rs[tid + off]; rq[tid] += rq[tid + off]; }
        __syncthreads();
    }
    const float mean = rs[0] * (1.0f / NPTS);
    const float var  = fmaxf(rq[0] * (1.0f / NPTS) - mean * mean, 0.0f);
    const float rs1  = rsqrtf(var + 1e-3f);            // inorm2d eps
    const float vart = var * rs1 * rs1;                // var after inorm (mean is 0)
    const float R    = rs1 * rsqrtf(vart + 1e-5f);     // + bnorm2d eps
    const float gg = g[c], bb = b[c];
    for (int i = tid; i < NPTS; i += WG)
        dst[i] = fmaxf((src[i] - mean) * R * gg + bb, 0.0f);
}

// ---------------------------------------------------------------------------
// Stage-2 norms on z = U[o,n] + V[o,m]:
//   mean = mU + mV, var = varU + varV (cross term is exactly 0).
// Emits P,Q transposed (position-major, 128 contiguous o per position) so the
// score kernel reads 512B contiguous runs.  bnorm beta folded into Q.
// ---------------------------------------------------------------------------
__global__ void __launch_bounds__(WG) stage2_norm_kernel(
    const float* __restrict__ U, const float* __restrict__ V,
    const float* __restrict__ g, const float* __restrict__ b,
    float* __restrict__ Pt, float* __restrict__ Qt)
{
    const int o   = blockIdx.x;      // 0..127
    const int tid = threadIdx.x;
    __shared__ float r0[WG], r1[WG], r2[WG], r3[WG];
    const float* Ur = U + (size_t)o * NPTS;
    const float* Vr = V + (size_t)o * NPTS;
    float sU = 0.f, qU = 0.f, sV = 0.f, qV = 0.f;
    for (int i = tid; i < NPTS; i += WG) {
        float a = Ur[i]; sU += a; qU += a * a;
        float c = Vr[i]; sV += c; qV += c * c;
    }
    r0[tid] = sU; r1[tid] = qU; r2[tid] = sV; r3[tid] = qV; __syncthreads();
    for (int off = WG / 2; off > 0; off >>= 1) {
        if (tid < off) {
            r0[tid] += r0[tid + off]; r1[tid] += r1[tid + off];
            r2[tid] += r2[tid + off]; r3[tid] += r3[tid + off];
        }
        __syncthreads();
    }
    const float mU = r0[0] * (1.0f / NPTS), mV = r2[0] * (1.0f / NPTS);
    const float vU = fmaxf(r1[0] * (1.0f / NPTS) - mU * mU, 0.f);
    const float vV = fmaxf(r3[0] * (1.0f / NPTS) - mV * mV, 0.f);
    const float var = vU + vV;
    const float rs1 = rsqrtf(var + 1e-3f);
    const float R   = rs1 * rsqrtf(var * rs1 * rs1 + 1e-5f);
    const float sg  = R * g[o];
    const float bt  = b[o];
    for (int i = tid; i < NPTS; i += WG) {
        Pt[(size_t)i * 128 + o] = (Ur[i] - mU) * sg;
        Qt[(size_t)i * 128 + o] = (Vr[i] - mV) * sg + bt;
    }
}

// ---------------------------------------------------------------------------
// Separable shot conv: s1[n] = <shot_w[0:64], q[:,n]>, s2[m] = <shot_w[64:], k[:,m]>
// ---------------------------------------------------------------------------
__global__ void __launch_bounds__(WG) shot_kernel(
    const float* __restrict__ q, const float* __restrict__ kx,
    const float* __restrict__ sw, float* __restrict__ s1, float* __restrict__ s2)
{
    const int pos = blockIdx.x * WG + threadIdx.x;
    if (pos >= NPTS) return;
    const int which = blockIdx.y;
    const float* X = which ? kx : q;
    const float* w = which ? (sw + C64) : sw;
    float acc = 0.f;
    for (int c = 0; c < C64; ++c)
        acc = fmaf(w[c], X[(size_t)c * NPTS + pos], acc);
    (which ? s2 : s1)[pos] = acc;
}

// ---------------------------------------------------------------------------
// Fused score + softmax (the only O(N^2 * 2C) kernel; never builds [128,N,M]):
//   score_pre[n,m] = sum_o w2[o]*relu(P[o,n]+Q[o,m]) + s1[n]+s2[m]+shot_b+b2
// One block per row n; each thread covers 3 m's; block softmax over 768.
// ---------------------------------------------------------------------------
__global__ void __launch_bounds__(WG) score_softmax_kernel(
    const float* __restrict__ Pt, const float* __restrict__ Qt,
    const float* __restrict__ w2, const float* __restrict__ s1,
    const float* __restrict__ s2, const float* __restrict__ shot_b,
    const float* __restrict__ b2, float* __restrict__ score)
{
    const int n   = blockIdx.x;
    const int tid = threadIdx.x;
    __shared__ __align__(16) float Psh[128];
    __shared__ __align__(16) float Wsh[128];
    __shared__ float red[WG];
    if (tid < 128) { Psh[tid] = Pt[(size_t)n * 128 + tid]; Wsh[tid] = w2[tid]; }
    __syncthreads();
    const float cbase = s1[n] + shot_b[0] + b2[0];
    float vals[3];
    float lmax = -3.4e38f;
    for (int j = 0; j < 3; ++j) {
        const int m = tid + j * WG;
        const float4* qv = reinterpret_cast<const float4*>(Qt + (size_t)m * 128);
        if (j < 2) __builtin_prefetch(Qt + (size_t)(m + WG) * 128, 0, 1);
        float acc = 0.f;
#pragma unroll
        for (int o4 = 0; o4 < 32; ++o4) {
            const float4 qq = qv[o4];
            const float4 pp = *reinterpret_cast<const float4*>(Psh + o4 * 4);
            const float4 ww = *reinterpret_cast<const float4*>(Wsh + o4 * 4);
            acc = fmaf(ww.x, fmaxf(pp.x + qq.x, 0.f), acc);
            acc = fmaf(ww.y, fmaxf(pp.y + qq.y, 0.f), acc);
            acc = fmaf(ww.z, fmaxf(pp.z + qq.z, 0.f), acc);
            acc = fmaf(ww.w, fmaxf(pp.w + qq.w, 0.f), acc);
        }
        const float sc = acc + cbase + s2[m];
        vals[j] = sc; lmax = fmaxf(lmax, sc);
    }
    red[tid] = lmax; __syncthreads();
    for (int off = WG / 2; off > 0; off >>= 1) {
        if (tid < off) red[tid] = fmaxf(red[tid], red[tid + off]);
        __syncthreads();
    }
    const float gmax = red[0]; __syncthreads();
    float esum = 0.f;
    for (int j = 0; j < 3; ++j) { vals[j] = __expf(vals[j] - gmax); esum += vals[j]; }
    red[tid] = esum; __syncthreads();
    for (int off = WG / 2; off > 0; off >>= 1) {
        if (tid < off) red[tid] += red[tid + off];
        __syncthreads();
    }
    const float inv = 1.0f / red[0];
    for (int j = 0; j < 3; ++j)
        score[(size_t)n * NPTS + tid + j * WG] = vals[j] * inv;
}

// bnorm1d(cat_bn) + relu, per channel of g1 [128,768]
__global__ void __launch_bounds__(WG) bn1d_relu_kernel(
    const float* __restrict__ X, const float* __restrict__ g,
    const float* __restrict__ b, float* __restrict__ Y)
{
    const int c   = blockIdx.x;
    const int tid = threadIdx.x;
    const float* src = X + (size_t)c * NPTS;
    float*       dst = Y + (size_t)c * NPTS;
    __shared__ float rs[WG], rq[WG];
    float s = 0.f, sq = 0.f;
    for (int i = tid; i < NPTS; i += WG) { float x = src[i]; s += x; sq += x * x; }
    rs[tid] = s; rq[tid] = sq; __syncthreads();
    for (int off = WG / 2; off > 0; off >>= 1) {
        if (tid < off) { rs[tid] += rs[tid + off]; rq[tid] += rq[tid + off]; }
        __syncthreads();
    }
    const float mean = rs[0] * (1.0f / NPTS);
    const float var  = fmaxf(rq[0] * (1.0f / NPTS) - mean * mean, 0.f);
    const float R    = rsqrtf(var + 1e-5f);
    const float gg = g[c], bb = b[c];
    for (int i = tid; i < NPTS; i += WG)
        dst[i] = fmaxf((src[i] - mean) * R * gg + bb, 0.f);
}

__global__ void __launch_bounds__(WG) cat_pack_kernel(
    const float* __restrict__ d1, const float* __restrict__ av2,
    float* __restrict__ cat)
{
    const int i = blockIdx.x * WG + threadIdx.x;
    const int half = C64 * NPTS;
    if (i >= 2 * half) return;
    cat[i] = (i < half) ? d1[i] : av2[i - half];
}

// ---------------------------------------------------------------------------
extern "C" void kernel_launch(void* const* d_in, const int* in_sizes, int n_in,
                              void* d_out, int out_size, void* d_ws, size_t ws_size,
                              hipStream_t stream)
{
    const float* desc1      = (const float*)d_in[0];
    const float* desc2      = (const float*)d_in[1];
    const float* qw         = (const float*)d_in[2];
    const float* qb         = (const float*)d_in[3];
    const float* kw         = (const float*)d_in[4];
    const float* kb         = (const float*)d_in[5];
    const float* vw         = (const float*)d_in[6];
    const float* vb         = (const float*)d_in[7];
    const float* mh_w       = (const float*)d_in[8];
    const float* mh_b       = (const float*)d_in[9];
    const float* cat_w1     = (const float*)d_in[10];
    const float* cat_b1     = (const float*)d_in[11];
    const float* cat_bn_g   = (const float*)d_in[12];
    const float* cat_bn_b   = (const float*)d_in[13];
    const float* cat_w2     = (const float*)d_in[14];
    const float* cat_b2     = (const float*)d_in[15];
    const float* sep_shot_w = (const float*)d_in[16];
    const float* sep_shot_b = (const float*)d_in[17];
    const float* sep_bn1_g  = (const float*)d_in[18];
    const float* sep_bn1_b  = (const float*)d_in[19];
    const float* sep_w1     = (const float*)d_in[20];
    const float* sep_b1     = (const float*)d_in[21];
    const float* sep_bn2_g  = (const float*)d_in[22];
    const float* sep_bn2_b  = (const float*)d_in[23];
    const float* sep_w2     = (const float*)d_in[24];
    const float* sep_b2     = (const float*)d_in[25];

    const int CN = C64 * NPTS;       // 49152
    float* ws  = (float*)d_ws;
    float* wq  = ws;                 // [64,768]
    float* wk  = wq  + CN;
    float* wv  = wk  + CN;
    float* Aq  = wv  + CN;           // stage-1 relu'd q channels
    float* Bk  = Aq  + CN;           // stage-1 relu'd k channels
    float* U   = Bk  + CN;           // [128,768]
    float* V   = U   + 2 * CN;
    float* Pt  = V   + 2 * CN;       // [768,128] position-major
    float* Qt  = Pt  + 2 * CN;
    float* s1  = Qt  + 2 * CN;       // [768]
    float* s2  = s1  + NPTS;
    float* av  = s2  + NPTS;         // [64,768]
    float* av2 = av  + CN;
    float* cat = av2 + CN;           // [128,768]
    float* g1  = cat + 2 * CN;
    float* hbn = g1  + 2 * CN;

    float* out   = (float*)d_out;    // desc1_new [64,768] first
    float* score = out + CN;         // then score [768,768]

    dim3 b32(32), bWG(WG);

    // q/k/v projections: W[64,64] @ X[64,768] + bias  (WMMA f32 16x16x4)
    wmma_gemm_kernel<0, true, false><<<dim3(NT16, 1), b32, 0, stream>>>(qw, 64, desc1, NPTS, qb, nullptr, 0, wq, NPTS, 64);
    wmma_gemm_kernel<0, true, false><<<dim3(NT16, 1), b32, 0, stream>>>(kw, 64, desc2, NPTS, kb, nullptr, 0, wk, NPTS, 64);
    wmma_gemm_kernel<0, true, false><<<dim3(NT16, 1), b32, 0, stream>>>(vw, 64, desc2, NPTS, vb, nullptr, 0, wv, NPTS, 64);

    stage1_norm_kernel<<<128, bWG, 0, stream>>>(wq, wk, sep_bn1_g, sep_bn1_b, Aq, Bk);
    shot_kernel<<<dim3((NPTS + WG - 1) / WG, 2), bWG, 0, stream>>>(wq, wk, sep_shot_w, s1, s2);

    // U = sep_w1[:, :64] @ Aq + b1 ; V = sep_w1[:, 64:] @ Bk
    wmma_gemm_kernel<0, true,  false><<<dim3(NT16, 2), b32, 0, stream>>>(sep_w1,      128, Aq, NPTS, sep_b1,  nullptr, 0, U, NPTS, 64);
    wmma_gemm_kernel<0, false, false><<<dim3(NT16, 2), b32, 0, stream>>>(sep_w1 + 64, 128, Bk, NPTS, nullptr, nullptr, 0, V, NPTS, 64);

    stage2_norm_kernel<<<128, bWG, 0, stream>>>(U, V, sep_bn2_g, sep_bn2_b, Pt, Qt);
    score_softmax_kernel<<<NPTS, bWG, 0, stream>>>(Pt, Qt, sep_w2, s1, s2, sep_shot_b, sep_b2, score);

    // add_value[d,n] = sum_m v[d,m] * score[n,m]   (B accessed transposed)
    wmma_gemm_kernel<1, false, false><<<dim3(NT16, 1), b32, 0, stream>>>(wv, NPTS, score, NPTS, nullptr, nullptr, 0, av, NPTS, NPTS);
    // mh projection
    wmma_gemm_kernel<0, true, false><<<dim3(NT16, 1), b32, 0, stream>>>(mh_w, 64, av, NPTS, mh_b, nullptr, 0, av2, NPTS, 64);

    cat_pack_kernel<<<(2 * CN + WG - 1) / WG, bWG, 0, stream>>>(desc1, av2, cat);
    wmma_gemm_kernel<0, true, false><<<dim3(NT16, 2), b32, 0, stream>>>(cat_w1, 128, cat, NPTS, cat_b1, nullptr, 0, g1, NPTS, 128);
    bn1d_relu_kernel<<<128, bWG, 0, stream>>>(g1, cat_bn_g, cat_bn_b, hbn);
    // desc1_new = desc1 + cat_w2 @ hbn + cat_b2   (residual fused in epilogue)
    wmma_gemm_kernel<0, true, true><<<dim3(NT16, 1), b32, 0, stream>>>(cat_w2, 128, hbn, NPTS, cat_b2, desc1, NPTS, out, NPTS, 128);
}